// Model_40080634806961
// MI455X (gfx1250) — compile-verified
//
#include <hip/hip_runtime.h>
#include <hip/hip_bf16.h>

// ---------------------------------------------------------------------------
// Hetero TransformerConv (machines<->jobs) + edge MLP head, fp32, CDNA5.
// Dense linears use V_WMMA_F32_16X16X4_F32 (native fp32 WMMA, exact fp32
// GEMM semantics). Shapes are templated so K-loops fully unroll and bounds
// guards become clamped-address loads + v_cndmask (no EXEC manipulation:
// WMMA requires EXEC all-ones). Edge softmax/aggregation: atomics.
// ---------------------------------------------------------------------------

typedef float v2f __attribute__((ext_vector_type(2)));
typedef float v8f __attribute__((ext_vector_type(8)));

// ---------------------------------------------------------------------------
// Y[n,DOUT] = act(X[n,DIN] @ W[DIN,DOUT] + b), one wave32 per 16x16 tile.
// A layout (16x4 f32): lanes 0-15 M=0..15 {K=k0,k0+1}; lanes 16-31 {K=k0+2,k0+3}
// B layout (4x16 f32): VGPR0 lanes0-15 K=k0 / lanes16-31 K=k0+2; VGPR1 K=k0+1/k0+3
// C/D layout: acc[r]: half0 -> row row0+r, half1 -> row row0+8+r, col=col0+(lane&15)
// ---------------------------------------------------------------------------
template <int DIN, int DOUT, int ACT>
__global__ __launch_bounds__(256) void lin_wmma_k(
    const float* __restrict__ X, int nrows,
    const float* __restrict__ W, const float* __restrict__ bias,
    float* __restrict__ Y)
{
    constexpr int NTN = (DOUT + 15) / 16;
    int wave = (blockIdx.x * blockDim.x + threadIdx.x) >> 5;
    int lane = threadIdx.x & 31;
    int ntm = (nrows + 15) >> 4;
    if (wave >= ntm * NTN) return;                 // wave-uniform branch
    int tm = wave / NTN, tn = wave - tm * NTN;
    int row0 = tm << 4, col0 = tn << 4;
    int half = lane >> 4, l = lane & 15;

    int  arow = row0 + l;
    bool rv   = arow < nrows;
    int  arc  = rv ? arow : (nrows - 1);           // clamped: address always valid
    int  bcol = col0 + l;
    bool cv   = (DOUT % 16 == 0) ? true : (bcol < DOUT);
    int  bcc  = cv ? bcol : (DOUT - 1);
    const float* Xr = X + arc * DIN;

    v8f acc = {};
#pragma unroll
    for (int k0 = 0; k0 < DIN; k0 += 4) {
        int ka = k0 + 2 * half;
        // clamp K when DIN isn't a multiple of 4 (address stays valid)
        int  kc0 = (DIN % 4 == 0) ? ka     : (ka     < DIN ? ka     : DIN - 1);
        int  kc1 = (DIN % 4 == 0) ? ka + 1 : (ka + 1 < DIN ? ka + 1 : DIN - 1);
        bool kv0 = (DIN % 4 == 0) || (ka     < DIN);
        bool kv1 = (DIN % 4 == 0) || (ka + 1 < DIN);
        float a0 = Xr[kc0];
        float a1 = Xr[kc1];
        float b0 = W[kc0 * DOUT + bcc];
        float b1 = W[kc1 * DOUT + bcc];
        v2f a, b;
        a.x = (rv && kv0) ? a0 : 0.0f;
        a.y = (rv && kv1) ? a1 : 0.0f;
        b.x = (cv && kv0) ? b0 : 0.0f;
        b.y = (cv && kv1) ? b1 : 0.0f;
        acc = __builtin_amdgcn_wmma_f32_16x16x4_f32(
            false, a, false, b, (short)0, acc, false, false);
    }

    float bv = bias ? bias[bcc] : 0.0f;
    int rbase = row0 + half * 8;
#pragma unroll
    for (int r = 0; r < 8; ++r) {
        int row = rbase + r;
        if (cv && row < nrows) {
            float o = acc[r] + bv;
            if (ACT) o = tanhf(o);
            Y[row * DOUT + bcol] = o;
        }
    }
}

// ---------------------------------------------------------------------------
// Head lin4 with fused gather: A[row,k] = concat(xm[src[row]](48), ea[row](4),
// xj[dst[row]](48)); hid[row,0:16] = tanh(A @ W4[100,16] + b4). K=100, dout=16.
// ---------------------------------------------------------------------------
__global__ __launch_bounds__(256) void head_lin4_wmma_k(
    const float* __restrict__ xm, const float* __restrict__ xj,
    const int* __restrict__ src, const int* __restrict__ dst,
    const float* __restrict__ ea, int E,
    const float* __restrict__ W4, const float* __restrict__ b4,
    float* __restrict__ hid)
{
    int wave = (blockIdx.x * blockDim.x + threadIdx.x) >> 5;
    int lane = threadIdx.x & 31;
    int ntm = (E + 15) >> 4;
    if (wave >= ntm) return;                       // wave-uniform
    int row0 = wave << 4;
    int half = lane >> 4, l = lane & 15;
    int  arow = row0 + l;
    bool rv   = arow < E;
    int  arc  = rv ? arow : (E - 1);               // clamped gather row
    int s = src[arc];
    int d = dst[arc];
    const float* xmr = xm + s * 48;
    const float* xjr = xj + d * 48;
    const float* ear = ea + arc * 4;

    v8f acc = {};
#pragma unroll
    for (int k0 = 0; k0 < 100; k0 += 4) {
        int k1 = k0 + 2 * half;
        int k2 = k1 + 1;
        float a0 = (k1 < 48) ? xmr[k1] : (k1 < 52 ? ear[k1 - 48] : xjr[k1 - 52]);
        float a1 = (k2 < 48) ? xmr[k2] : (k2 < 52 ? ear[k2 - 48] : xjr[k2 - 52]);
        v2f a, b;
        a.x = rv ? a0 : 0.0f;
        a.y = rv ? a1 : 0.0f;
        b.x = W4[k1 * 16 + l];
        b.y = W4[k2 * 16 + l];
        acc = __builtin_amdgcn_wmma_f32_16x16x4_f32(
            false, a, false, b, (short)0, acc, false, false);
    }

    float bv = b4[l];
    int rbase = row0 + half * 8;
#pragma unroll
    for (int r = 0; r < 8; ++r) {
        int row = rbase + r;
        if (row < E) hid[row * 16 + l] = tanhf(acc[r] + bv);
    }
}

// lin3 (16->1) + mask
__global__ void head_lin3_k(const float* __restrict__ hid,
                            const float* __restrict__ W3, const float* __restrict__ b3,
                            const int* __restrict__ mask, float* __restrict__ out, int E)
{
    int e = blockIdx.x * blockDim.x + threadIdx.x;
    if (e >= E) return;
    float acc = b3[0];
    const float* h = hid + e * 16;
#pragma unroll
    for (int j = 0; j < 16; ++j) acc += h[j] * W3[j];
    out[e] = (mask[e] != 0) ? acc : 0.0f;
}

// --------- segment-softmax attention, edge-parallel kernels ----------------
__device__ __forceinline__ unsigned f32_to_key(float f) {
    unsigned u = __float_as_uint(f);
    return (u >> 31) ? ~u : (u | 0x80000000u);
}
__device__ __forceinline__ float key_to_f32(unsigned k) {
    return (k >> 31) ? __uint_as_float(k & 0x7FFFFFFFu) : __uint_as_float(~k);
}

// scores: a[e,h] = 0.25 * sum_c q[dst,h,c]*(k[src,h,c] + (ea@We)[h,c]); seg-max
__global__ void score_k(const float* __restrict__ q, const float* __restrict__ kk,
                        const int* __restrict__ src, const int* __restrict__ dst,
                        const float* __restrict__ ea, const float* __restrict__ We,
                        float* __restrict__ sc, unsigned* __restrict__ amax, int E)
{
    int e = blockIdx.x * blockDim.x + threadIdx.x;
    if (e >= E) return;
    int s = src[e], d = dst[e];
    float e0 = ea[e * 4 + 0], e1 = ea[e * 4 + 1];
    float e2 = ea[e * 4 + 2], e3 = ea[e * 4 + 3];
    const float* qr = q  + d * 48;
    const float* kr = kk + s * 48;
#pragma unroll
    for (int h = 0; h < 3; ++h) {
        float a = 0.0f;
#pragma unroll
        for (int c = 0; c < 16; ++c) {
            int hc = h * 16 + c;
            float ehc = e0 * We[hc] + e1 * We[48 + hc] + e2 * We[96 + hc] + e3 * We[144 + hc];
            a += qr[hc] * (kr[hc] + ehc);
        }
        a *= 0.25f;                                // 1/sqrt(C), C=16
        sc[e * 3 + h] = a;
        atomicMax(&amax[d * 3 + h], f32_to_key(a));
    }
}

// ex = exp(a - amax[dst]); denom[dst] += ex  (sc overwritten with ex)
__global__ void expsum_k(float* __restrict__ sc, const unsigned* __restrict__ amax,
                         float* __restrict__ denom, const int* __restrict__ dst, int E)
{
    int idx = blockIdx.x * blockDim.x + threadIdx.x;
    if (idx >= E * 3) return;
    int e = idx / 3, h = idx - e * 3;
    int d = dst[e];
    float m = key_to_f32(amax[d * 3 + h]);
    float ex = expf(sc[idx] - m);
    sc[idx] = ex;
    atomicAdd(&denom[d * 3 + h], ex);
}

// agg[dst,hc] += (v[src,hc] + (ea@We)[hc]) * ex/denom[dst,h]
__global__ void message_k(const float* __restrict__ vv, const float* __restrict__ sc,
                          const float* __restrict__ denom,
                          const int* __restrict__ src, const int* __restrict__ dst,
                          const float* __restrict__ ea, const float* __restrict__ We,
                          float* __restrict__ agg, int E)
{
    int idx = blockIdx.x * blockDim.x + threadIdx.x;
    if (idx >= E * 48) return;
    int e = idx / 48, hc = idx - e * 48;
    int h = hc >> 4;
    int s = src[e], d = dst[e];
    float ehc = ea[e * 4 + 0] * We[hc]      + ea[e * 4 + 1] * We[48 + hc]
              + ea[e * 4 + 2] * We[96 + hc] + ea[e * 4 + 3] * We[144 + hc];
    float alpha = sc[e * 3 + h] / denom[d * 3 + h];
    atomicAdd(&agg[d * 48 + hc], (vv[s * 48 + hc] + ehc) * alpha);
}

// x = tanh(agg + skip)
__global__ void addskip_tanh_k(float* __restrict__ agg, const float* __restrict__ sk, int n)
{
    int i = blockIdx.x * blockDim.x + threadIdx.x;
    if (i < n) agg[i] = tanhf(agg[i] + sk[i]);
}

// ---------------------------------------------------------------------------
// Host orchestration
// ---------------------------------------------------------------------------
struct ConvP { const float *We, *Wk, *bk, *Wq, *bq, *Wsk, *bsk, *Wv, *bv; };

extern "C" void kernel_launch(void* const* d_in, const int* in_sizes, int n_in,
                              void* d_out, int out_size, void* d_ws, size_t ws_size,
                              hipStream_t stream)
{
    // --- input map ---------------------------------------------------------
    // Top level in setup_inputs() insertion order; nested `params` pytree
    // flattened JAX-style (dict keys sorted alphabetically at every level):
    //   params: convs < lin1_j < lin1_m < lin3 < lin4
    //   convs[l]: j2m < m2j ; conv dict: e < k < q < skip < v ; lin: W < b
    const float* x_machine = (const float*)d_in[0];
    const float* x_job     = (const float*)d_in[1];
    const int*   ei_src    = (const int*)d_in[2];
    const int*   ei_dst    = (const int*)d_in[3];
    const float* edge_attr = (const float*)d_in[4];
    const float* edge_rev  = (const float*)d_in[5];
    const int*   mask      = (const int*)d_in[6];

    auto F = [&](int i) { return (const float*)d_in[i]; };
    auto getConv = [&](int base) {
        ConvP p;
        p.We  = F(base + 0);                        // e.W  [4,48]
        p.Wk  = F(base + 1); p.bk  = F(base + 2);   // k    [din,48],[48]
        p.Wq  = F(base + 3); p.bq  = F(base + 4);   // q
        p.Wsk = F(base + 5); p.bsk = F(base + 6);   // skip
        p.Wv  = F(base + 7); p.bv  = F(base + 8);   // v
        return p;
    };
    ConvP c0_j2m = getConv(7);
    ConvP c0_m2j = getConv(16);
    ConvP c1_j2m = getConv(25);
    ConvP c1_m2j = getConv(34);
    const float* W1j = F(43); const float* b1j = F(44);   // lin1_j [12,8],[8]
    const float* W1m = F(45); const float* b1m = F(46);   // lin1_m [6,8],[8]
    const float* W3  = F(47); const float* b3  = F(48);   // lin3  [16,1],[1]
    const float* W4  = F(49); const float* b4  = F(50);   // lin4  [100,16],[16]

    const int E  = in_sizes[2];
    const int NM = in_sizes[0] / 6;
    const int NJ = in_sizes[1] / 12;
    float* out = (float*)d_out;

    // --- workspace partition (fp32 words) ---------------------------------
    float* ws = (float*)d_ws;
    size_t o = 0;
    float* xm_a = ws + o; o += (size_t)NM * 48;
    float* xj_a = ws + o; o += (size_t)NJ * 48;
    float* xm_b = ws + o; o += (size_t)NM * 48;
    float* xj_b = ws + o; o += (size_t)NJ * 48;
    float* qb   = ws + o; o += (size_t)NJ * 48;
    float* kb   = ws + o; o += (size_t)NJ * 48;
    float* vb   = ws + o; o += (size_t)NJ * 48;
    float* skb  = ws + o; o += (size_t)NJ * 48;
    float* sc   = ws + o; o += (size_t)E * 3;
    unsigned* amax = (unsigned*)(ws + o); o += (size_t)NJ * 3;
    float* denom = ws + o; o += (size_t)NJ * 3;
    float* hid  = ws + o; o += (size_t)E * 16;
    (void)ws_size; (void)n_in; (void)out_size;

    auto blocks_for = [](int n, int ntn) {
        int tiles = ((n + 15) / 16) * ntn;
        return (tiles + 7) / 8;                    // 8 waves per 256-thread block
    };
    // node linear into 48 cols, din is 8 (layer 0) or 48 (layer 1)
    auto lin48 = [&](const float* X, int n, int din, const float* W, const float* b,
                     float* Y) {
        if (din == 8)
            lin_wmma_k<8, 48, 0><<<blocks_for(n, 3), 256, 0, stream>>>(X, n, W, b, Y);
        else
            lin_wmma_k<48, 48, 0><<<blocks_for(n, 3), 256, 0, stream>>>(X, n, W, b, Y);
    };

    auto conv_dir = [&](const float* srcx, const float* dstx, int n_src, int n_dst,
                        const int* sidx, const int* didx, const float* eatt,
                        const ConvP& p, int din, float* aggout) {
        hipMemsetAsync(aggout, 0, (size_t)n_dst * 48 * sizeof(float), stream);
        hipMemsetAsync(amax,   0, (size_t)n_dst * 3 * sizeof(unsigned), stream);
        hipMemsetAsync(denom,  0, (size_t)n_dst * 3 * sizeof(float), stream);
        lin48(dstx, n_dst, din, p.Wq,  p.bq,  qb);
        lin48(srcx, n_src, din, p.Wk,  p.bk,  kb);
        lin48(srcx, n_src, din, p.Wv,  p.bv,  vb);
        lin48(dstx, n_dst, din, p.Wsk, p.bsk, skb);
        score_k  <<<(E + 255) / 256, 256, 0, stream>>>(qb, kb, sidx, didx, eatt, p.We, sc, amax, E);
        expsum_k <<<(E * 3 + 255) / 256, 256, 0, stream>>>(sc, amax, denom, didx, E);
        message_k<<<(E * 48 + 255) / 256, 256, 0, stream>>>(vb, sc, denom, sidx, didx, eatt, p.We, aggout, E);
        addskip_tanh_k<<<((n_dst * 48) + 255) / 256, 256, 0, stream>>>(aggout, skb, n_dst * 48);
    };

    // lin1 + tanh per node type (widths 6->8, 12->8)
    lin_wmma_k<6, 8, 1><<<blocks_for(NM, 1), 256, 0, stream>>>(x_machine, NM, W1m, b1m, xm_a);
    lin_wmma_k<12, 8, 1><<<blocks_for(NJ, 1), 256, 0, stream>>>(x_job, NJ, W1j, b1j, xj_a);

    // layer 0 (din=8): both convs read (xm_a, xj_a), write (_b)
    conv_dir(xm_a, xj_a, NM, NJ, ei_src, ei_dst, edge_attr, c0_m2j, 8, xj_b);
    conv_dir(xj_a, xm_a, NJ, NM, ei_dst, ei_src, edge_rev,  c0_j2m, 8, xm_b);
    // layer 1 (din=48): read (_b), write (_a)
    conv_dir(xm_b, xj_b, NM, NJ, ei_src, ei_dst, edge_attr, c1_m2j, 48, xj_a);
    conv_dir(xj_b, xm_b, NJ, NM, ei_dst, ei_src, edge_rev,  c1_j2m, 48, xm_a);

    // edge scoring head: concat-gather fused into WMMA lin4, then lin3+mask
    {
        int blocks = (((E + 15) / 16) + 7) / 8;
        head_lin4_wmma_k<<<blocks, 256, 0, stream>>>(xm_a, xj_a, ei_src, ei_dst,
                                                     edge_attr, E, W4, b4, hid);
    }
    head_lin3_k<<<(E + 255) / 256, 256, 0, stream>>>(hid, W3, b3, mask, out, E);
}